// MRTE_25503515803978
// MI455X (gfx1250) — compile-verified
//
#include <hip/hip_runtime.h>

// LengthRegulator: out[b,t,:] = x[b, searchsorted(cumsum(dur[b]), t, 'right'), :]
// zero-padded for t >= total duration. Pure data movement; HBM-write bound.

typedef float v4f __attribute__((ext_vector_type(4)));

#define LR_B 16
#define LR_L 512
#define LR_D 512
#define LR_T 4096

#define THREADS 256
#define WAVES (THREADS / 32)
#define FRAMES_PER_BLOCK 64
#define QPR (LR_D / 4 / 32)   // float4s per lane per 512-float row = 4

#if defined(__gfx1250__)
typedef __attribute__((address_space(1))) int* glb_i32_ptr;
typedef __attribute__((address_space(3))) int* lds_i32_ptr;
#endif

__global__ __launch_bounds__(THREADS)
void lr_gather_kernel(const float* __restrict__ x,
                      const int* __restrict__ dur,
                      float* __restrict__ out)
{
    __shared__ int s_ends[LR_L];

    const int b   = blockIdx.y;
    const int t0  = blockIdx.x * FRAMES_PER_BLOCK;
    const int tid = threadIdx.x;

    const int* __restrict__ durb = dur + b * LR_L;

    // ---- stage this batch's durations into LDS ------------------------------
#if defined(__gfx1250__) && __has_builtin(__builtin_amdgcn_global_load_async_to_lds_b32)
    // CDNA5 async global->LDS DMA path (tracked by ASYNCcnt)
    for (int i = tid; i < LR_L; i += THREADS) {
        __builtin_amdgcn_global_load_async_to_lds_b32(
            (glb_i32_ptr)(durb + i),
            (lds_i32_ptr)(&s_ends[i]),
            /*offset=*/0, /*cpol=*/0);
    }
#if __has_builtin(__builtin_amdgcn_s_wait_asynccnt)
    __builtin_amdgcn_s_wait_asynccnt(0);
#else
    asm volatile("s_wait_asynccnt 0" ::: "memory");
#endif
#else
    for (int i = tid; i < LR_L; i += THREADS) s_ends[i] = durb[i];
#endif
    __syncthreads();

    // ---- inclusive scan: s_ends[i] = sum_{k<=i} dur[k]  (Hillis-Steele) -----
    const int i0 = tid;
    const int i1 = tid + THREADS;
    for (int off = 1; off < LR_L; off <<= 1) {
        int a0 = (i0 >= off) ? s_ends[i0 - off] : 0;
        int a1 = (i1 >= off) ? s_ends[i1 - off] : 0;
        __syncthreads();            // all reads done before any writes
        s_ends[i0] += a0;
        s_ends[i1] += a1;
        __syncthreads();
    }

    const int total = s_ends[LR_L - 1];

    const v4f* __restrict__ xb = (const v4f*)(x + (size_t)b * LR_L * LR_D);
    v4f* __restrict__ ob = (v4f*)(out + ((size_t)b * LR_T + t0) * (size_t)LR_D);

    const int lane = tid & 31;
    const int wave = tid >> 5;

    // Each wave copies whole 2KiB rows: 32 lanes x 4 x b128.
    for (int f = wave; f < FRAMES_PER_BLOCK; f += WAVES) {
        const int t = t0 + f;
        v4f* __restrict__ orow = ob + (size_t)f * (LR_D / 4);

        if (t < total) {
            // first j with ends[j] > t  (uniform across the wave -> LDS broadcast)
            int lo = 0, hi = LR_L;
            while (lo < hi) {
                int mid = (lo + hi) >> 1;
                if (s_ends[mid] > t) hi = mid; else lo = mid + 1;
            }
            const v4f* __restrict__ xrow = xb + (size_t)lo * (LR_D / 4);
#pragma unroll
            for (int k = 0; k < QPR; ++k) {
                v4f v = xrow[lane + 32 * k];                 // RT load: keep x in L2
                __builtin_nontemporal_store(v, &orow[lane + 32 * k]);  // NT: stream out
            }
        } else {
            v4f z = {0.f, 0.f, 0.f, 0.f};
#pragma unroll
            for (int k = 0; k < QPR; ++k)
                __builtin_nontemporal_store(z, &orow[lane + 32 * k]);
        }
    }
}

extern "C" void kernel_launch(void* const* d_in, const int* in_sizes, int n_in,
                              void* d_out, int out_size, void* d_ws, size_t ws_size,
                              hipStream_t stream)
{
    (void)in_sizes; (void)n_in; (void)d_ws; (void)ws_size; (void)out_size;

    const float* x   = (const float*)d_in[0];
    const int*   dur = (const int*)d_in[1];
    // d_in[2] = mel_max_length (device scalar) == LR_T, fixed by the reference.
    float* out = (float*)d_out;

    dim3 grid(LR_T / FRAMES_PER_BLOCK, LR_B, 1);
    lr_gather_kernel<<<grid, THREADS, 0, stream>>>(x, dur, out);
}